// QuantumLayer_24859270709748
// MI455X (gfx1250) — compile-verified
//
#include <hip/hip_runtime.h>
#include <math.h>

// MI455X / gfx1250 state-vector quantum circuit simulator.
//  - layer = low-side 256x256 complex GEMM (in-place, LDS tile) +
//            high-side 256x256 complex GEMM fused with CNOT-ring scatter
//  - complex GEMM on V_WMMA_F32_16X16X4_F32 (fp32 matrix cores)
//  - state/gate panels staged into LDS by the Tensor Data Mover
//    (tensor_load_to_lds + s_wait_tensorcnt), with TDM LDS padding used to
//    produce bank-conflict-free row strides (258 / 24 floats).
//  - LDS reads go through laundered address_space(3) pointers so they lower
//    to ds_load_* (DScnt only), not flat_load_*.
// Workspace: 4 state planes (re/im x 2 buffers) + 4 gate planes ~= 537 MB.

typedef float v2f __attribute__((ext_vector_type(2)));
typedef float v8f __attribute__((ext_vector_type(8)));
typedef unsigned int u32x4 __attribute__((ext_vector_type(4)));
typedef int i32x4 __attribute__((ext_vector_type(4)));
typedef int i32x8 __attribute__((ext_vector_type(8)));
typedef __attribute__((address_space(3))) float lds_f;

#define NQ    16
#define NL    8
#define BATCH 512
#define DIM   65536   // 2^16 amplitudes
#define HDIM  256     // 2^8 sub-dimension
#define LDS_STRIDE 258  // low_gemm LDS row stride (256 + TDM pad of 2 DWORDs)
#define PANEL_STRIDE 24 // high_gemm LDS row stride (16 + TDM pad of 8 DWORDs)

__device__ __forceinline__ v8f wmma_f32(v2f a, v2f b, v8f c) {
  // (neg_a, A, neg_b, B, c_mod, C, reuse_a, reuse_b); f32 has no A/B neg.
  return __builtin_amdgcn_wmma_f32_16x16x4_f32(false, a, false, b, (short)0, c,
                                               false, false);
}

// ---- Tensor Data Mover: 2D tile load Global -> LDS (data_size = 4 bytes) ----
// g1w0 carries data_size + LDS padding controls (D# group1 word0).
// This toolchain uses the 6-arg builtin: (g0, g1, g2, g3, g_extra, cpol).
__device__ __forceinline__ void tdm_load_2d(unsigned lds_off,
                                            unsigned long long gaddr,
                                            unsigned g1w0,
                                            unsigned dim0, unsigned dim1,
                                            unsigned tile0, unsigned tile1,
                                            unsigned stride0) {
  u32x4 g0;
  g0[0] = 1u;                                   // count=1, user descriptor
  g0[1] = lds_off;                              // lds_addr [63:32]
  g0[2] = (unsigned)(gaddr & 0xffffffffu);      // global_addr [95:64]
  g0[3] = (unsigned)((gaddr >> 32) & 0x01ffffffu) | 0x80000000u; // addr hi + type=2
  i32x8 g1;
  g1[0] = (int)g1w0;                            // wg_mask=0, data_size, padding
  g1[1] = (int)(dim0 << 16);                    // tensor_dim0 lo16 -> [63:48]
  g1[2] = (int)((dim0 >> 16) | (dim1 << 16));   // tensor_dim0 hi / tensor_dim1 lo
  g1[3] = (int)((dim1 >> 16) | (tile0 << 16));  // tensor_dim1 hi / tile_dim0
  g1[4] = (int)tile1;                           // tile_dim1 (tile_dim2 = 0)
  g1[5] = (int)stride0;                         // tensor_dim0_stride lo32
  g1[6] = 0;                                    // stride0 hi / dim1_stride lo
  g1[7] = 0;
  i32x4 z4 = {0, 0, 0, 0};
  i32x8 z8 = {0, 0, 0, 0, 0, 0, 0, 0};
  __builtin_amdgcn_tensor_load_to_lds(g0, g1, z4, z4, z8, 0);
}

// data_size=4B (code 2); pad every 256 DWORDs (code 7) by 2 DWORDs (code 1)
#define G1W0_ROW256 ((2u << 16) | (1u << 20) | (7u << 22) | (1u << 25))
// data_size=4B; pad every 16 DWORDs (code 3) by 8 DWORDs (code 7)
#define G1W0_ROW16  ((2u << 16) | (1u << 20) | (3u << 22) | (7u << 25))

__device__ __forceinline__ unsigned lds_off_of(const void* p) {
  // generic LDS address: aperture in [63:32], byte offset in [31:0]
  return (unsigned)(unsigned long long)(uintptr_t)p;
}

// CNOT ring of layer with shift r: sequential i=0..15, control wire i,
// target wire (i+r)%16; wire w <-> bit (15-w) of flat index j.
__device__ __forceinline__ unsigned cnot_perm(unsigned j, int r) {
  #pragma unroll
  for (int i = 0; i < 16; ++i) {
    int tw = (i + r) & 15;
    unsigned cb = 15u - (unsigned)i;
    unsigned tb = 15u - (unsigned)tw;
    j ^= ((j >> cb) & 1u) << tb;
  }
  return j;
}

// ---------------- init: product state from x rotations ----------------
__global__ void init_state(const float* __restrict__ x,
                           float* __restrict__ sre, float* __restrict__ sim) {
  __shared__ float ca[16], sa[16];
  const int b = blockIdx.y;
  if (threadIdx.x < 16) {
    float a = 0.5f * 3.14159265358979323846f * x[b * NQ + threadIdx.x];
    ca[threadIdx.x] = cosf(a);
    sa[threadIdx.x] = sinf(a);
  }
  __syncthreads();
  unsigned j = blockIdx.x * blockDim.x + threadIdx.x;  // 0..65535
  float p = 1.f;
  #pragma unroll
  for (int i = 0; i < 16; ++i)
    p *= ((j >> (15 - i)) & 1u) ? sa[i] : ca[i];
  size_t off = (size_t)b * DIM + j;
  sre[off] = p;
  sim[off] = 0.f;
}

// ---------------- per-layer gate matrices (Kronecker products) ----------------
// Glo[k=l_old][n=l_new] = prod_{i=8..15} R_i[bit_{15-i}(n), bit_{15-i}(k)]
// Ghi[m=h_new][k=h_old] = prod_{i=0..7}  R_i[bit_{7-i}(m),  bit_{7-i}(k)]
__global__ void build_gates(const float* __restrict__ w,   // [16][3] this layer
                            float* __restrict__ glo_re, float* __restrict__ glo_im,
                            float* __restrict__ ghi_re, float* __restrict__ ghi_im) {
  __shared__ float Rre[16][2][2], Rim[16][2][2];
  const int t = threadIdx.x;
  if (t < 16) {
    float phi = w[t * 3 + 0], th = w[t * 3 + 1], om = w[t * 3 + 2];
    float c = cosf(0.5f * th), s = sinf(0.5f * th);
    float ap = 0.5f * (phi + om), am = 0.5f * (phi - om);
    float epr = cosf(ap), epi = -sinf(ap);   // ep = exp(-0.5i(phi+om))
    float emr = cosf(am), emi = -sinf(am);   // em = exp(-0.5i(phi-om))
    Rre[t][0][0] = epr * c;  Rim[t][0][0] = epi * c;    // m00 = ep*c
    Rre[t][0][1] = -emr * s; Rim[t][0][1] = emi * s;    // m01 = -conj(em)*s
    Rre[t][1][0] = emr * s;  Rim[t][1][0] = emi * s;    // m10 = em*s
    Rre[t][1][1] = epr * c;  Rim[t][1][1] = -epi * c;   // m11 = conj(ep)*c
  }
  __syncthreads();
  unsigned e = blockIdx.x * blockDim.x + t;  // 0..65535
  unsigned p = e >> 8, q = e & 255u;
  {
    float re = 1.f, im = 0.f;
    #pragma unroll
    for (int i = 8; i < 16; ++i) {
      int sh = 15 - i;
      int bn = (q >> sh) & 1, bo = (p >> sh) & 1;
      float rr = Rre[i][bn][bo], ri = Rim[i][bn][bo];
      float nr = re * rr - im * ri;
      im = re * ri + im * rr; re = nr;
    }
    glo_re[e] = re; glo_im[e] = im;
  }
  {
    float re = 1.f, im = 0.f;
    #pragma unroll
    for (int i = 0; i < 8; ++i) {
      int sh = 7 - i;
      int bn = (p >> sh) & 1, bo = (q >> sh) & 1;
      float rr = Rre[i][bn][bo], ri = Rim[i][bn][bo];
      float nr = re * rr - im * ri;
      im = re * ri + im * rr; re = nr;
    }
    ghi_re[e] = re; ghi_im[e] = im;
  }
}

// ---------------- low-side complex GEMM, in-place ----------------
// rows = (b,h) pairs; out[row, n] = sum_k S[row, k] * Glo[k, n]
// 16x256 tile staged into LDS by the TDM (padded to 258-float rows).
__global__ void low_gemm(float* __restrict__ sre, float* __restrict__ sim,
                         const float* __restrict__ gre, const float* __restrict__ gim) {
  __shared__ float Sre[16 * LDS_STRIDE];
  __shared__ float Sim[16 * LDS_STRIDE];
  const int t = threadIdx.x;
  const size_t row0 = (size_t)blockIdx.x * 16;
  if (t < 32) {  // wave 0 issues the DMA; TDM ignores EXEC -> one issue
    unsigned long long gr = (unsigned long long)(uintptr_t)(sre + row0 * HDIM);
    unsigned long long gi = (unsigned long long)(uintptr_t)(sim + row0 * HDIM);
    tdm_load_2d(lds_off_of(Sre), gr, G1W0_ROW256, 256, 16, 256, 16, 256);
    tdm_load_2d(lds_off_of(Sim), gi, G1W0_ROW256, 256, 16, 256, 16, 256);
    __builtin_amdgcn_s_wait_tensorcnt(0);
  }
  __syncthreads();
  // launder LDS pointers (arrays written only by the TDM) as address_space(3)
  // so reads lower to ds_load_* instead of flat_load_*.
  lds_f* lre = (lds_f*)Sre;
  lds_f* lim = (lds_f*)Sim;
  asm volatile("" : "+v"(lre), "+v"(lim));
  const int wave  = t >> 5;
  const int lane  = t & 31;
  const int m     = lane & 15;
  const int khalf = lane >> 4;
  const int kh    = khalf << 1;  // +0 or +2 within each K=4 group
  #pragma unroll
  for (int half = 0; half < 2; ++half) {
    const int n0 = (wave + half * 8) * 16;
    const int n  = n0 + m;   // this lane's B/D column
    v8f acc_re = {}; v8f acc_im = {};
    for (int k = 0; k < 256; k += 4) {
      int kk = k + kh;
      v2f a_re, a_im, b_re, b_im, b_imn;
      a_re.x = lre[m * LDS_STRIDE + kk];     a_re.y = lre[m * LDS_STRIDE + kk + 1];
      a_im.x = lim[m * LDS_STRIDE + kk];     a_im.y = lim[m * LDS_STRIDE + kk + 1];
      b_re.x = gre[kk * HDIM + n];           b_re.y = gre[(kk + 1) * HDIM + n];
      b_im.x = gim[kk * HDIM + n];           b_im.y = gim[(kk + 1) * HDIM + n];
      b_imn.x = -b_im.x; b_imn.y = -b_im.y;
      acc_re = wmma_f32(a_re, b_re,  acc_re);
      acc_re = wmma_f32(a_im, b_imn, acc_re);
      acc_im = wmma_f32(a_re, b_im,  acc_im);
      acc_im = wmma_f32(a_im, b_re,  acc_im);
    }
    #pragma unroll
    for (int r = 0; r < 8; ++r) {
      int ml = r + 8 * khalf;
      size_t g = (row0 + ml) * HDIM + (n0 + m);
      sre[g] = acc_re[r];
      sim[g] = acc_im[r];
    }
  }
}

// ---------------- high-side complex GEMM + CNOT permutation scatter ----------------
// out[b, hn, l] = sum_k Ghi[hn, k] * S[b, k, l]; store to j' = cnot_perm(hn<<8|l).
// Block owns one 16-wide l-panel (staged once by TDM, strided gather) and
// computes all 16 h-tiles with its 8 waves (2 tiles per wave).
__global__ void high_gemm_scatter(const float* __restrict__ sre, const float* __restrict__ sim,
                                  float* __restrict__ dre, float* __restrict__ dimg,
                                  const float* __restrict__ are, const float* __restrict__ aim,
                                  int rshift) {
  __shared__ float Lre[HDIM * PANEL_STRIDE];
  __shared__ float Lim[HDIM * PANEL_STRIDE];
  const int b  = blockIdx.y;
  const int t  = threadIdx.x;
  const int l0 = blockIdx.x * 16;
  const size_t base = (size_t)b * DIM;
  if (t < 32) {  // wave 0 issues the strided panel gathers
    unsigned long long gr = (unsigned long long)(uintptr_t)(sre + base + l0);
    unsigned long long gi = (unsigned long long)(uintptr_t)(sim + base + l0);
    tdm_load_2d(lds_off_of(Lre), gr, G1W0_ROW16, 16, 256, 16, 256, 256);
    tdm_load_2d(lds_off_of(Lim), gi, G1W0_ROW16, 16, 256, 16, 256, 256);
    __builtin_amdgcn_s_wait_tensorcnt(0);
  }
  __syncthreads();
  lds_f* lre = (lds_f*)Lre;
  lds_f* lim = (lds_f*)Lim;
  asm volatile("" : "+v"(lre), "+v"(lim));
  const int wave  = t >> 5;
  const int lane  = t & 31;
  const int m     = lane & 15;
  const int khalf = lane >> 4;
  const int kh    = khalf << 1;
  #pragma unroll
  for (int half = 0; half < 2; ++half) {
    const int h0 = (wave + half * 8) * 16;
    v8f acc_re = {}; v8f acc_im = {};
    for (int k = 0; k < 256; k += 4) {
      int kk = k + kh;
      v2f a_re, a_im, b_re, b_im, b_imn;
      a_re.x = are[(h0 + m) * HDIM + kk];    a_re.y = are[(h0 + m) * HDIM + kk + 1];
      a_im.x = aim[(h0 + m) * HDIM + kk];    a_im.y = aim[(h0 + m) * HDIM + kk + 1];
      b_re.x = lre[kk * PANEL_STRIDE + m];   b_re.y = lre[(kk + 1) * PANEL_STRIDE + m];
      b_im.x = lim[kk * PANEL_STRIDE + m];   b_im.y = lim[(kk + 1) * PANEL_STRIDE + m];
      b_imn.x = -b_im.x; b_imn.y = -b_im.y;
      acc_re = wmma_f32(a_re, b_re,  acc_re);
      acc_re = wmma_f32(a_im, b_imn, acc_re);
      acc_im = wmma_f32(a_re, b_im,  acc_im);
      acc_im = wmma_f32(a_im, b_re,  acc_im);
    }
    #pragma unroll
    for (int r = 0; r < 8; ++r) {
      unsigned hn = (unsigned)(h0 + r + 8 * khalf);
      unsigned l  = (unsigned)(l0 + m);
      unsigned j  = cnot_perm((hn << 8) | l, rshift);
      dre[base + j]  = acc_re[r];
      dimg[base + j] = acc_im[r];
    }
  }
}

// ---------------- Z expectation values ----------------
__global__ void expect_kernel(const float* __restrict__ sre, const float* __restrict__ sim,
                              float* __restrict__ out) {
  const int b = blockIdx.x;
  const int t = threadIdx.x;
  const size_t base = (size_t)b * DIM;
  float acc[16];
  #pragma unroll
  for (int i = 0; i < 16; ++i) acc[i] = 0.f;
  for (int chunk = 0; chunk < 256; ++chunk) {
    unsigned j = (unsigned)(chunk * 256 + t);
    float re = sre[base + j], im = sim[base + j];
    float p = re * re + im * im;
    #pragma unroll
    for (int i = 0; i < 16; ++i)
      acc[i] += ((j >> (15 - i)) & 1u) ? -p : p;
  }
  __shared__ float red[256];
  for (int i = 0; i < 16; ++i) {
    red[t] = acc[i];
    __syncthreads();
    for (int s = 128; s > 0; s >>= 1) {
      if (t < s) red[t] += red[t + s];
      __syncthreads();
    }
    if (t == 0) out[b * NQ + i] = red[0];
    __syncthreads();
  }
}

extern "C" void kernel_launch(void* const* d_in, const int* in_sizes, int n_in,
                              void* d_out, int out_size, void* d_ws, size_t ws_size,
                              hipStream_t stream) {
  const float* x = (const float*)d_in[0];        // [512,16]
  const float* w = (const float*)d_in[1];        // [8,16,3]
  float* out = (float*)d_out;                    // [512,16]
  const size_t N = (size_t)BATCH * DIM;
  float* ws = (float*)d_ws;
  float* s0re = ws;
  float* s0im = ws + N;
  float* s1re = ws + 2 * N;
  float* s1im = ws + 3 * N;
  float* glo_re = ws + 4 * N;
  float* glo_im = glo_re + DIM;
  float* ghi_re = glo_im + DIM;
  float* ghi_im = ghi_re + DIM;

  init_state<<<dim3(DIM / 256, BATCH), 256, 0, stream>>>(x, s0re, s0im);

  float *cre = s0re, *cim = s0im, *ore = s1re, *oim = s1im;
  for (int l = 0; l < NL; ++l) {
    build_gates<<<DIM / 256, 256, 0, stream>>>(w + l * NQ * 3,
                                               glo_re, glo_im, ghi_re, ghi_im);
    low_gemm<<<(BATCH * HDIM) / 16, 256, 0, stream>>>(cre, cim, glo_re, glo_im);
    int r = l % (NQ - 1) + 1;
    high_gemm_scatter<<<dim3(16, BATCH), 256, 0, stream>>>(cre, cim, ore, oim,
                                                           ghi_re, ghi_im, r);
    float* tmp;
    tmp = cre; cre = ore; ore = tmp;
    tmp = cim; cim = oim; oim = tmp;
  }
  expect_kernel<<<BATCH, 256, 0, stream>>>(cre, cim, out);
}